// ExpertParallel_63711544868877
// MI455X (gfx1250) — compile-verified
//
#include <hip/hip_runtime.h>
#include <hip/hip_bf16.h>

// MoE expert-parallel dispatch+combine, single-process form.
//
// The reference's sort -> all-to-all(identity) -> inverse-scatter pipeline is
// an exact identity permutation: output[n, k, :] = x[n, :] for k in {0,1},
// independent of expert_indices. So the kernel is pure data movement:
// duplicate each 16KB row of x into two consecutive 16KB output rows.
//
// Roofline: 128 MB read + 256 MB write = 384 MB @ 23.3 TB/s => ~16.5 us floor.
// Zero FLOPs -> no WMMA. The CDNA5-optimal path is the async global<->LDS DMA
// (GLOBAL_LOAD_ASYNC_TO_LDS_B128 / GLOBAL_STORE_ASYNC_FROM_LDS_B128, ASYNCcnt):
// no VGPR payload, deep async queues, and the x2 duplication is done by issuing
// two async stores from the same LDS bytes. NT hints keep L2 unpolluted
// (every byte is touched exactly once on each side).

#define N_TOKENS   8192
#define HIDDEN     4096
#define F4_PER_ROW (HIDDEN / 4)       // 1024 float4 per row (16 KB)
#define BLOCK      256                // 8 waves (wave32)
#define PER_THREAD (F4_PER_ROW / BLOCK) // 4

__global__ __launch_bounds__(BLOCK)
void ExpertParallel_rowdup_async(const float4* __restrict__ x,
                                 float4* __restrict__ out)
{
    // One 16 KB row staged in LDS per block. lbuf starts at LDS offset 0;
    // low 32 bits of the flat address of an LDS object are its LDS byte
    // address (ISA: LDS aperture truncates to addr[31:0]).
    __shared__ float4 lbuf[F4_PER_ROW];

    const uint32_t row = blockIdx.x;
    const uint32_t tid = threadIdx.x;

    const float4* __restrict__ src  = x   + (size_t)row * F4_PER_ROW;
    float4* __restrict__       dst0 = out + (size_t)row * (2 * F4_PER_ROW);
    float4* __restrict__       dst1 = dst0 + F4_PER_ROW;

    // Phase 1: async DMA global -> LDS (row staged once). 4 x b128 per lane,
    // lanes of a wave cover 512B contiguous per instruction (fully coalesced).
#pragma unroll
    for (int j = 0; j < PER_THREAD; ++j) {
        const uint32_t col     = tid + j * BLOCK;
        const uint32_t lds_off = (uint32_t)(uintptr_t)&lbuf[col];
        const float4*  g       = src + col;
        asm volatile("global_load_async_to_lds_b128 %0, %1, off th:TH_LOAD_NT"
                     :: "v"(lds_off), "v"(g)
                     : "memory");
    }

    // Async LDS writes are unordered vs. subsequent async LDS reads within the
    // wave -> must drain ASYNCcnt before the stores read the staged bytes.
    // Each lane only touches its own LDS slots, so no workgroup barrier needed.
    asm volatile("s_wait_asynccnt 0x0" ::: "memory");

    // Phase 2: async DMA LDS -> global, same LDS bytes stored to BOTH
    // destination rows (the k=0 / k=1 duplicates). Non-temporal: the 256 MB
    // output stream is never re-read.
#pragma unroll
    for (int j = 0; j < PER_THREAD; ++j) {
        const uint32_t col     = tid + j * BLOCK;
        const uint32_t lds_off = (uint32_t)(uintptr_t)&lbuf[col];
        float4*        g0      = dst0 + col;
        float4*        g1      = dst1 + col;
        asm volatile("global_store_async_from_lds_b128 %0, %1, off th:TH_STORE_NT"
                     :: "v"(g0), "v"(lds_off)
                     : "memory");
        asm volatile("global_store_async_from_lds_b128 %0, %1, off th:TH_STORE_NT"
                     :: "v"(g1), "v"(lds_off)
                     : "memory");
    }

    // Drain before wave exit (S_ENDPGM does an implicit wait-idle, but be
    // explicit so LDS reuse by a successor workgroup on this WGP is safe).
    asm volatile("s_wait_asynccnt 0x0" ::: "memory");
}

extern "C" void kernel_launch(void* const* d_in, const int* in_sizes, int n_in,
                              void* d_out, int out_size, void* d_ws, size_t ws_size,
                              hipStream_t stream) {
    (void)in_sizes; (void)n_in; (void)out_size; (void)d_ws; (void)ws_size;

    const float4* x   = (const float4*)d_in[0];  // [8192, 4096] f32
    // d_in[1] (expert_indices) and d_in[2] (ep_world_size) provably do not
    // affect the output (the permutation cancels itself) -> unused.
    float4* out = (float4*)d_out;                // [8192, 2, 4096] f32

    ExpertParallel_rowdup_async<<<N_TOKENS, BLOCK, 0, stream>>>(x, out);
}